// BareboneDRGCN_30786325577792
// MI455X (gfx1250) — compile-verified
//
#include <hip/hip_runtime.h>
#include <hip/hip_bf16.h>

// ---------------- problem constants (match reference) ----------------
#define N_NODES 100000
#define IN_C    64
#define OUT_C   64
#define NUM_REL 8
#define R2      16          // 2*NUM_REL
#define NUM_BASES 36
#define NT_NODES (N_NODES / 16)   // 6250 exactly

#define SZ_SUM ((size_t)R2 * N_NODES * IN_C)   // 102,400,000 floats
#define SZ_CNT ((size_t)R2 * N_NODES)          //   1,600,000 floats
#define SZ_W   ((size_t)R2 * IN_C * OUT_C)     //      65,536 floats

typedef float v2f __attribute__((ext_vector_type(2)));
typedef float v8f __attribute__((ext_vector_type(8)));

// ---------------- kernel 0: zero the accumulators --------------------
__global__ void drgcn_zero(float4* __restrict__ p, long long n4) {
    long long i = (long long)blockIdx.x * blockDim.x + threadIdx.x;
    long long stride = (long long)gridDim.x * blockDim.x;
    float4 z; z.x = 0.f; z.y = 0.f; z.z = 0.f; z.w = 0.f;
    for (; i < n4; i += stride) p[i] = z;
}

// ---------------- kernel 1: W[r] = (mask*comp)[r] @ weight -----------
__global__ void drgcn_build_w(const float* __restrict__ comp,
                              const float* __restrict__ weight,
                              float* __restrict__ Wout) {
    int r = blockIdx.x;               // 0..15
    __shared__ float coef[NUM_BASES];
    if (threadIdx.x < NUM_BASES) {
        int b = threadIdx.x;
        float m = 0.f;
        if (b < 4) {
            m = 1.f;                                   // NUM_M shared bases
        } else if (b < 20) {
            int col = (b - 4) >> 1;                    // NUM_N=2 per relation
            if ((r & 7) == col) m = 1.f;               // r==col or r==col+8
        } else {
            int col = b - 20;                          // NUM_O=1 per R2
            if (r == col) m = 1.f;
        }
        coef[b] = m * comp[r * NUM_BASES + b];
    }
    __syncthreads();
    for (int i = threadIdx.x; i < IN_C * OUT_C; i += blockDim.x) {
        float s = 0.f;
        #pragma unroll 4
        for (int b = 0; b < NUM_BASES; ++b)
            s += coef[b] * weight[(size_t)b * (IN_C * OUT_C) + i];
        Wout[(size_t)r * (IN_C * OUT_C) + i] = s;
    }
}

// ---------------- kernel 2: edge scatter (sum + count) ---------------
// 16 threads per edge, 4 channels each; handles both edge directions.
__global__ void drgcn_scatter(const float* __restrict__ x,
                              const int*   __restrict__ ei,   // [2, E] flat
                              const int*   __restrict__ et,   // [E]
                              float* __restrict__ sum,
                              float* __restrict__ cnt,
                              int nEdges) {
    long long gid = (long long)blockIdx.x * blockDim.x + threadIdx.x;
    int e = (int)(gid >> 4);
    if (e >= nEdges) return;
    int t = (int)(gid & 15);
    int c = t * 4;

    int src = ei[e];
    int dst = ei[nEdges + e];
    int rel = et[e];

    // forward edge: (rel, dst) <- x[src]
    {
        const float4 v = *(const float4*)(x + (size_t)src * IN_C + c);
        float* p = sum + (((size_t)rel * N_NODES + dst) * IN_C + c);
        atomicAdd(p + 0, v.x); atomicAdd(p + 1, v.y);
        atomicAdd(p + 2, v.z); atomicAdd(p + 3, v.w);
    }
    // reversed edge: (rel+NUM_REL, src) <- x[dst]
    {
        const float4 v = *(const float4*)(x + (size_t)dst * IN_C + c);
        float* p = sum + ((((size_t)rel + NUM_REL) * N_NODES + src) * IN_C + c);
        atomicAdd(p + 0, v.x); atomicAdd(p + 1, v.y);
        atomicAdd(p + 2, v.z); atomicAdd(p + 3, v.w);
    }
    if (t == 0) {
        atomicAdd(cnt + (size_t)rel * N_NODES + dst, 1.0f);
        atomicAdd(cnt + ((size_t)rel + NUM_REL) * N_NODES + src, 1.0f);
    }
}

// ---------------- kernel 3: WMMA GEMM --------------------------------
// out[n,o] = sum_r (sum[r,n,:]/max(cnt[r,n],1)) @ W[r][:,o] + x[n,:]@root[:,o] + bias[o]
// One wave32 per 16-node tile. fp32 WMMA 16x16x4, K stepped by 4.
// A layout (fp32 16x4): lane&15 = M row; float2 = K{kk,kk+1}, kk = k0 + 2*(lane>>4)
// C/D layout: lane&15 = N col; VGPR v = M row v + 8*(lane>>4)
__global__ void __launch_bounds__(256)
drgcn_gemm(const float* __restrict__ sum,
           const float* __restrict__ cnt,
           const float* __restrict__ W,
           const float* __restrict__ x,
           const float* __restrict__ root,
           const float* __restrict__ bias,
           float* __restrict__ out) {
    int wave = (int)((blockIdx.x * blockDim.x + threadIdx.x) >> 5);
    int lane = threadIdx.x & 31;
    if (wave >= NT_NODES) return;                 // uniform per wave

    const int mrow  = lane & 15;                  // M index held by this lane
    const int khalf = lane >> 4;                  // 0 or 1 (K sub-pair select)
    const int row   = wave * 16 + mrow;           // global node row

    v8f acc[4];
    #pragma unroll
    for (int nt = 0; nt < 4; ++nt) acc[nt] = (v8f)0.0f;

    // 17 K-panels: r = 0..15 relations (scaled A), r = 16 root (A = x)
    for (int r = 0; r < 17; ++r) {
        const float* Asrc;
        const float* Bsrc;
        float scale;
        if (r < 16) {
            Asrc  = sum + ((size_t)r * N_NODES + row) * IN_C;
            float c = cnt[(size_t)r * N_NODES + row];
            scale = 1.0f / fmaxf(c, 1.0f);
            Bsrc  = W + (size_t)r * (IN_C * OUT_C);
        } else {
            Asrc  = x + (size_t)row * IN_C;
            scale = 1.0f;
            Bsrc  = root;
        }
        for (int k0 = 0; k0 < IN_C; k0 += 4) {
            const int kk = k0 + 2 * khalf;
            v2f a = *(const v2f*)(Asrc + kk);
            a.x *= scale; a.y *= scale;
            const float* brow = Bsrc + (size_t)kk * OUT_C;
            #pragma unroll
            for (int nt = 0; nt < 4; ++nt) {
                const int col = nt * 16 + mrow;
                v2f b;
                b.x = brow[col];
                b.y = brow[OUT_C + col];
                acc[nt] = __builtin_amdgcn_wmma_f32_16x16x4_f32(
                    /*neg_a=*/false, a, /*neg_b=*/false, b,
                    /*c_mod=*/(short)0, acc[nt],
                    /*reuse_a=*/false, /*reuse_b=*/false);
            }
        }
    }

    // store D + bias
    #pragma unroll
    for (int nt = 0; nt < 4; ++nt) {
        const int col = nt * 16 + mrow;
        const float bc = bias[col];
        #pragma unroll
        for (int v = 0; v < 8; ++v) {
            int orow = wave * 16 + v + 8 * khalf;
            out[(size_t)orow * OUT_C + col] = acc[nt][v] + bc;
        }
    }
}

// ---------------- launcher ------------------------------------------
extern "C" void kernel_launch(void* const* d_in, const int* in_sizes, int n_in,
                              void* d_out, int out_size, void* d_ws, size_t ws_size,
                              hipStream_t stream) {
    const float* x      = (const float*)d_in[0];
    const int*   ei     = (const int*)d_in[1];
    const int*   et     = (const int*)d_in[2];
    const float* weight = (const float*)d_in[3];
    const float* comp   = (const float*)d_in[4];
    const float* root   = (const float*)d_in[5];
    const float* bias   = (const float*)d_in[6];
    float* out = (float*)d_out;

    float* ws  = (float*)d_ws;
    float* sum = ws;
    float* cnt = ws + SZ_SUM;
    float* Wd  = ws + SZ_SUM + SZ_CNT;

    const int nEdges = in_sizes[2];               // 2,000,000

    // 1) zero sum + cnt (contiguous, multiple of 4 floats)
    {
        long long n4 = (long long)(SZ_SUM + SZ_CNT) / 4;
        drgcn_zero<<<8192, 256, 0, stream>>>((float4*)sum, n4);
    }
    // 2) per-relation weights
    drgcn_build_w<<<R2, 256, 0, stream>>>(comp, weight, Wd);
    // 3) edge scatter (both directions)
    {
        long long threads = (long long)nEdges * 16;
        int blocks = (int)((threads + 255) / 256);
        drgcn_scatter<<<blocks, 256, 0, stream>>>(x, ei, et, sum, cnt, nEdges);
    }
    // 4) fused mean + 17-panel WMMA GEMM + bias
    {
        int waves_per_block = 256 / 32;
        int blocks = (NT_NODES + waves_per_block - 1) / waves_per_block;
        drgcn_gemm<<<blocks, 256, 0, stream>>>(sum, cnt, Wd, x, root, bias, out);
    }
}